// CubicBSplineFFD_34900904247317
// MI455X (gfx1250) — compile-verified
//
#include <hip/hip_runtime.h>
#include <hip/hip_bf16.h>

// Cubic B-spline FFD upsampling, fused single pass.
// out[ch][x][y][z] = sum_{i,j,k} v[ch][i][j][k] * Wx(x,i)*Wy(y,j)*Wz(z,k)
// with W(a,b) = B3(|a + 5 - 5b| / 5)  (stride-5 transposed conv + crop folded in).
//
// One wave32 per 16x16x16 output tile.  Control tile (8x8x8) is brought into
// LDS by the Tensor Data Mover (tensor_load_to_lds, TENSORcnt) with hardware
// zero-fill past the 42^3 grid edge; the three separable contractions run on
// V_WMMA_F32_16X16X4_F32 through LDS staging.

typedef float v2f __attribute__((ext_vector_type(2)));
typedef float v8f __attribute__((ext_vector_type(8)));
typedef unsigned int v4u __attribute__((ext_vector_type(4)));
typedef int v4i __attribute__((ext_vector_type(4)));
typedef int v8i __attribute__((ext_vector_type(8)));

#define IMG 192
#define CPG 42
#define NCH 12   // 4 * NDIM(3)

__device__ __forceinline__ float bsw(int j, int i) {
    // cubic B-spline weight of control point i on output sample j (same axis).
    // |arg|/5 >= 2  <=>  outside the 19-tap kernel -> returns 0 automatically.
    float u = fabsf((float)(j + 5 - 5 * i)) * 0.2f;
    float r = 0.0f;
    if (u < 1.0f) {
        r = 0.6666666666666666f + (0.5f * u - 1.0f) * u * u;
    } else if (u < 2.0f) {
        float q = 2.0f - u;
        r = q * q * q * 0.16666666666666666f;
    }
    return r;
}

__global__ __launch_bounds__(32)
void ffd_upsample_wmma(const float* __restrict__ v, float* __restrict__ out) {
    const int ch = blockIdx.x / 12;   // n*3 + dim channel, 0..11
    const int bx = blockIdx.x % 12;   // 16-wide x block
    const int by = blockIdx.y;        // 16-wide y block
    const int bz = blockIdx.z;        // 16-wide z block

    const int lane = threadIdx.x;     // 0..31 (wave32)
    const int ln   = lane & 15;
    const int lh   = lane >> 4;       // lane half: 0 or 1

    const int j0x = bx * 16, j0y = by * 16, j0z = bz * 16;
    // first control point index whose support touches this block:
    // i0 = ceil((16b - 4)/5) == floor(16b/5); window of 8 covers all taps.
    const int i0x = (16 * bx) / 5;
    const int i0y = (16 * by) / 5;
    const int i0z = (16 * bz) / 5;

    __shared__ float lC[8 * 8 * 8];        // control tile [kx][ky][kz]
    __shared__ float lW[3 * 8 * 16];       // weight blocks [axis][k][n]  (x,y,z)
    __shared__ float lU1[64 * 16];         // stage-1 result [(kx*8+ky)][z]
    __shared__ float lU2[8 * 16 * 16];     // stage-2 result [kx][y][z]

    const float* vc = v + (size_t)ch * (CPG * CPG * CPG);
    __builtin_prefetch(vc + ((size_t)i0x * CPG + i0y) * CPG + i0z, 0, 0);

    // ---- control tile via Tensor Data Mover: 8x8x8 tile of a 42^3 tensor ----
    // dim0 = z (stride 1), dim1 = y (stride 42), dim2 = x (stride 1764).
    // tensor_dimN = remaining extent from tile origin -> OOB reads return 0,
    // which is exactly the transposed-conv zero padding at the high edge.
#if defined(__has_builtin) && __has_builtin(__builtin_amdgcn_tensor_load_to_lds)
    {
        const unsigned long long gaddr =
            (unsigned long long)(const void*)(vc + ((size_t)i0x * (CPG * CPG) +
                                                    (size_t)i0y * CPG + i0z));
        const unsigned int lds_addr = (unsigned int)(unsigned long long)&lC[0];
        const unsigned int td0 = (unsigned int)(CPG - i0z);   // z extent left
        const unsigned int td1 = (unsigned int)(CPG - i0y);   // y extent left
        const unsigned int td2 = (unsigned int)(CPG - i0x);   // x extent left
        const unsigned int s0 = CPG;             // dim0 stride (elements)
        const unsigned int s1 = CPG * CPG;       // dim1 stride (elements)

        v4u g0;
        g0[0] = 1u;                                      // count=1, no gather
        g0[1] = lds_addr;                                // lds_addr[31:0]
        g0[2] = (unsigned int)gaddr;                     // global_addr[31:0]
        g0[3] = (unsigned int)((gaddr >> 32) & 0x1FFFFFFu) | (2u << 30); // type=2

        v8i g1;
        g1[0] = (int)(2u << 16);                         // wg_mask=0, data_size=4B
        g1[1] = (int)((td0 & 0xFFFFu) << 16);            // tensor_dim0[15:0]
        g1[2] = (int)((td0 >> 16) | ((td1 & 0xFFFFu) << 16)); // td0 hi | td1 lo
        g1[3] = (int)((td1 >> 16) | (8u << 16));         // td1 hi | tile_dim0=8
        g1[4] = (int)(8u | (8u << 16));                  // tile_dim1=8 | tile_dim2=8
        g1[5] = (int)s0;                                 // tensor_dim0_stride lo32
        g1[6] = (int)((s1 & 0xFFFFu) << 16);             // str0 hi16=0 | str1 lo16
        g1[7] = (int)(s1 >> 16);                         // tensor_dim1_stride hi
        v4i g2;
        g2[0] = (int)td2;                                // tensor_dim2
        g2[1] = 0;                                       // tensor_dim3 (unused)
        g2[2] = (int)(CPG * CPG * CPG);                  // tensor_dim2_stride lo
        g2[3] = 0;                                       // str hi | tile_dim3=0
        v4i g3 = {0, 0, 0, 0};                           // no dim 4

#if __clang_major__ >= 23
        v8i gx = {0, 0, 0, 0, 0, 0, 0, 0};
        __builtin_amdgcn_tensor_load_to_lds(g0, g1, g2, g3, gx, 0);
#else
        __builtin_amdgcn_tensor_load_to_lds(g0, g1, g2, g3, 0);
#endif
    }
#else
    // fallback: per-lane gather with manual zero padding
    #pragma unroll
    for (int q = 0; q < 16; ++q) {
        int e  = lane + 32 * q;            // 0..511
        int kx = e >> 6, ky = (e >> 3) & 7, kz = e & 7;
        int gx = i0x + kx, gy = i0y + ky, gz = i0z + kz;
        float val = 0.0f;
        if (gx < CPG && gy < CPG && gz < CPG)
            val = vc[(gx * CPG + gy) * CPG + gz];
        lC[e] = val;
    }
#endif

    // ---- build the three 8x16 weight blocks (overlaps with the TDM DMA) ----
    #pragma unroll
    for (int q = 0; q < 12; ++q) {
        int e  = lane + 32 * q;            // 0..383
        int ax = e >> 7;                   // 0=x, 1=y, 2=z
        int kn = e & 127;
        int k = kn >> 4, n = kn & 15;
        int j0 = (ax == 0) ? j0x : (ax == 1) ? j0y : j0z;
        int i0 = (ax == 0) ? i0x : (ax == 1) ? i0y : i0z;
        lW[e] = bsw(j0 + n, i0 + k);
    }

#if defined(__has_builtin) && __has_builtin(__builtin_amdgcn_s_wait_tensorcnt)
    __builtin_amdgcn_s_wait_tensorcnt(0);   // TDM tile landed in LDS
#endif
    __syncthreads();

    const float* Wx = lW;                  // [k][x]
    const float* Wy = lW + 128;            // [k][y]
    const float* Wz = lW + 256;            // [k][z]

    // ===== stage 1: contract z.  U1[r=(kx*8+ky)][z] = sum_kz C[r][kz]*Wz[kz][z]
    // M=64 rows (4 WMMA M-tiles), K=8 (2 steps), N=16.
    #pragma unroll
    for (int t = 0; t < 4; ++t) {
        v8f acc = {0.f, 0.f, 0.f, 0.f, 0.f, 0.f, 0.f, 0.f};
        #pragma unroll
        for (int kk = 0; kk < 2; ++kk) {
            int k0 = kk * 4 + 2 * lh;          // K index of A/B element 0
            int r  = t * 16 + ln;              // M row
            v2f a, b;
            a[0] = lC[r * 8 + k0];
            a[1] = lC[r * 8 + k0 + 1];
            b[0] = Wz[k0 * 16 + ln];
            b[1] = Wz[(k0 + 1) * 16 + ln];
            acc = __builtin_amdgcn_wmma_f32_16x16x4_f32(
                false, a, false, b, (short)0, acc, false, false);
        }
        #pragma unroll
        for (int vr = 0; vr < 8; ++vr)         // D: VGPR vr -> M = vr + 8*lh
            lU1[(t * 16 + vr + 8 * lh) * 16 + ln] = acc[vr];
    }
    __syncthreads();

    // ===== stage 2: contract y.  U2[kx][y][z] = sum_ky Wy[ky][y]*U1[kx*8+ky][z]
    #pragma unroll
    for (int kx = 0; kx < 8; ++kx) {
        v8f acc = {0.f, 0.f, 0.f, 0.f, 0.f, 0.f, 0.f, 0.f};
        #pragma unroll
        for (int kk = 0; kk < 2; ++kk) {
            int k0 = kk * 4 + 2 * lh;
            v2f a, b;
            a[0] = Wy[k0 * 16 + ln];           // A[M=y][K=ky] = Wy^T
            a[1] = Wy[(k0 + 1) * 16 + ln];
            b[0] = lU1[(kx * 8 + k0) * 16 + ln];
            b[1] = lU1[(kx * 8 + k0 + 1) * 16 + ln];
            acc = __builtin_amdgcn_wmma_f32_16x16x4_f32(
                false, a, false, b, (short)0, acc, false, false);
        }
        #pragma unroll
        for (int vr = 0; vr < 8; ++vr)
            lU2[kx * 256 + (vr + 8 * lh) * 16 + ln] = acc[vr];
    }
    __syncthreads();

    // ===== stage 3: contract x.  out[x][y][z] = sum_kx Wx[kx][x]*U2[kx][y][z]
    float* outc = out + (size_t)ch * IMG * IMG * IMG;
    #pragma unroll
    for (int y = 0; y < 16; ++y) {
        v8f acc = {0.f, 0.f, 0.f, 0.f, 0.f, 0.f, 0.f, 0.f};
        #pragma unroll
        for (int kk = 0; kk < 2; ++kk) {
            int k0 = kk * 4 + 2 * lh;
            v2f a, b;
            a[0] = Wx[k0 * 16 + ln];
            a[1] = Wx[(k0 + 1) * 16 + ln];
            b[0] = lU2[k0 * 256 + y * 16 + ln];
            b[1] = lU2[(k0 + 1) * 256 + y * 16 + ln];
            acc = __builtin_amdgcn_wmma_f32_16x16x4_f32(
                false, a, false, b, (short)0, acc, false, false);
        }
        #pragma unroll
        for (int vr = 0; vr < 8; ++vr) {
            int X = j0x + vr + 8 * lh;
            outc[(size_t)X * (IMG * IMG) + (size_t)(j0y + y) * IMG + (j0z + ln)] =
                acc[vr];
        }
    }
}

extern "C" void kernel_launch(void* const* d_in, const int* in_sizes, int n_in,
                              void* d_out, int out_size, void* d_ws, size_t ws_size,
                              hipStream_t stream) {
    (void)in_sizes; (void)n_in; (void)out_size; (void)d_ws; (void)ws_size;
    const float* v = (const float*)d_in[0];     // (4,3,42,42,42) fp32
    float* out = (float*)d_out;                 // (4,3,192,192,192) fp32
    dim3 grid(NCH * 12, 12, 12);                // ch*xblk, yblk, zblk
    dim3 block(32);                             // one wave32 per tile
    hipLaunchKernelGGL(ffd_upsample_wmma, grid, block, 0, stream, v, out);
}